// MultiAgentMHCModel_13950053777923
// MI455X (gfx1250) — compile-verified
//
#include <hip/hip_runtime.h>
#include <hip/hip_bf16.h>

typedef __attribute__((ext_vector_type(2))) float v2f;
typedef __attribute__((ext_vector_type(8))) float v8f;
typedef __attribute__((ext_vector_type(4))) int   v4i;

#define DH   256
#define NAG  64
#define TM   64
#define TN   128

// ---- CDNA5 async global->LDS staging (ASYNCcnt path), guarded ----------------
#if defined(__HIP_DEVICE_COMPILE__) && defined(__gfx1250__) && \
    __has_builtin(__builtin_amdgcn_global_load_async_to_lds_b128) && \
    __has_builtin(__builtin_amdgcn_s_wait_asynccnt)
#define USE_ASYNC_LDS 1
#endif

#ifdef USE_ASYNC_LDS
typedef __attribute__((address_space(1))) v4i gv4i;   // global 128-bit chunk
typedef __attribute__((address_space(3))) v4i lv4i;   // LDS   128-bit chunk
#endif

__device__ __forceinline__ void copy16_g2l(void* lds_dst, const void* g_src) {
#ifdef USE_ASYNC_LDS
    __builtin_amdgcn_global_load_async_to_lds_b128((gv4i*)g_src, (lv4i*)lds_dst, 0, 0);
#else
    *(float4*)lds_dst = *(const float4*)g_src;
#endif
}

__device__ __forceinline__ void wait_g2l() {
#ifdef USE_ASYNC_LDS
    __builtin_amdgcn_s_wait_asynccnt(0);
#endif
}

// ---- WMMA f32 16x16x4 fragment helpers (wave32) ------------------------------
// A (16x4, MxK): lanes 0-15 -> M=lane, VGPR0=K0 / VGPR1=K1 ; lanes 16-31 -> M=lane-16, VGPR0=K2 / VGPR1=K3
__device__ __forceinline__ v2f frag_a(const float* __restrict__ p, int ld, int mbase, int kk, int lane) {
    int off = (lane >= 16) ? 2 : 0;
    int m = mbase + (lane & 15);
    v2f a;
    a.x = p[m * ld + kk + off];
    a.y = p[m * ld + kk + off + 1];
    return a;
}
// B (4x16, KxN): lanes 0-15 -> N=lane, VGPR0=B[K0][n] / VGPR1=B[K1][n]; lanes 16-31 -> K2/K3
__device__ __forceinline__ v2f frag_b(const float* __restrict__ p, int ld, int nbase, int kk, int lane) {
    int off = (lane >= 16) ? 2 : 0;
    int n = nbase + (lane & 15);
    v2f b;
    b.x = p[(kk + off)     * ld + n];
    b.y = p[(kk + off + 1) * ld + n];
    return b;
}
#define WMMA_F32(acc, a, b) \
    __builtin_amdgcn_wmma_f32_16x16x4_f32(false, (a), false, (b), (short)0, (acc), false, false)

// ---- Generic C = A[M,256] @ W[256,N] + bias, 64x128 block per workgroup ------
__global__ __launch_bounds__(256) void gemm_bias_kernel(const float* __restrict__ A,
                                                        const float* __restrict__ W,
                                                        const float* __restrict__ bias,
                                                        float* __restrict__ C,
                                                        int N) {
    __shared__ float Alds[TM * DH];   //  64 KB
    __shared__ float Wlds[DH * TN];   // 128 KB
    const int tid  = threadIdx.x;
    const int lane = tid & 31;
    const int wave = tid >> 5;
    const long m0 = (long)blockIdx.x * TM;
    const int  n0 = blockIdx.y * TN;

    // stage A block (64x256) and W block (256x128) via async global->LDS
    {
        const float* Ag = A + m0 * DH;
        for (int i = tid; i < TM * DH / 4; i += 256)
            copy16_g2l(&Alds[i * 4], Ag + i * 4);
        for (int i = tid; i < DH * TN / 4; i += 256) {
            int r  = i / (TN / 4);
            int c4 = i % (TN / 4);
            copy16_g2l(&Wlds[i * 4], W + (long)r * N + n0 + c4 * 4);
        }
        wait_g2l();
    }
    __syncthreads();

    const int tm  = wave >> 1;        // 0..3
    const int tnb = (wave & 1) * 4;   // 0 or 4
    for (int t = 0; t < 4; ++t) {
        const int tn = tnb + t;
        v8f acc = {};
        #pragma unroll 8
        for (int kk = 0; kk < DH; kk += 4) {
            v2f a = frag_a(Alds, DH, tm * 16, kk, lane);
            v2f b = frag_b(Wlds, TN, tn * 16, kk, lane);
            acc = WMMA_F32(acc, a, b);
        }
        const int colL  = n0 + tn * 16 + (lane & 15);
        const float bsv = bias[colL];
        const int rbase = tm * 16 + ((lane >= 16) ? 8 : 0);
        #pragma unroll
        for (int j = 0; j < 8; ++j)
            C[(m0 + rbase + j) * N + colL] = acc[j] + bsv;
    }
}

// ---- Fully fused MHC layer: one workgroup per batch element ------------------
__global__ __launch_bounds__(256) void mhc_layer_kernel(float* __restrict__ h,   // [B*64,256] in/out
                                                        const float* __restrict__ ln_g,
                                                        const float* __restrict__ ln_b,
                                                        const float* __restrict__ Wq, const float* __restrict__ bq,
                                                        const float* __restrict__ Wk, const float* __restrict__ bk,
                                                        const float* __restrict__ Wv, const float* __restrict__ bv,
                                                        const float* __restrict__ ab,  // [64,256]
                                                        const float* __restrict__ sf)  // [64]
{
    __shared__ float hl[NAG * DH];      // 64 KB  (layernormed h)
    __shared__ float ql[NAG * DH];      // 64 KB  (q, later reused for attn@v)
    __shared__ float kl[NAG * DH];      // 64 KB
    __shared__ float vl[NAG * DH];      // 64 KB
    __shared__ float sc[NAG * 65];      // 16.25 KB (padded scores / log-alpha / attn)

    const int tid  = threadIdx.x;
    const int lane = tid & 31;
    const int wave = tid >> 5;
    float* hg = h + (size_t)blockIdx.x * NAG * DH;

    // ---- load h block (async global->LDS) ----
    for (int i = tid; i < NAG * DH / 4; i += 256)
        copy16_g2l(&hl[i * 4], hg + i * 4);
    wait_g2l();
    __syncthreads();

    // ---- LayerNorm (in place): 4 lanes per row, 64 elems each ----
    {
        const int row = tid >> 2, q4 = tid & 3;
        float s = 0.f, s2 = 0.f;
        for (int j = 0; j < 64; ++j) {
            float x = hl[row * DH + q4 * 64 + j];
            s += x; s2 += x * x;
        }
        s  += __shfl_xor(s, 1);  s  += __shfl_xor(s, 2);
        s2 += __shfl_xor(s2, 1); s2 += __shfl_xor(s2, 2);
        const float mu = s * (1.f / 256.f);
        const float rs = rsqrtf(s2 * (1.f / 256.f) - mu * mu + 1e-5f);
        for (int j = 0; j < 64; ++j) {
            int c = q4 * 64 + j;
            hl[row * DH + c] = (hl[row * DH + c] - mu) * rs * ln_g[c] + ln_b[c];
        }
    }
    __syncthreads();

    // ---- q/k/v = LN(h) @ W + b : 192 16x16 tiles over 8 waves, W streamed from L2 ----
    {
        const int off = (lane >= 16) ? 2 : 0;
        for (int i = 0; i < 24; ++i) {
            const int g   = wave + 8 * i;       // 0..191
            const int mat = g >> 6;             // 0=q 1=k 2=v
            const int tt  = g & 63;
            const int tm  = tt >> 4, tn = tt & 15;
            const float* Wm = (mat == 0) ? Wq : (mat == 1) ? Wk : Wv;
            const float* bm = (mat == 0) ? bq : (mat == 1) ? bk : bv;
            float* dst      = (mat == 0) ? ql : (mat == 1) ? kl : vl;
            const int ncol = tn * 16 + (lane & 15);
            v8f acc = {};
            for (int kk = 0; kk < DH; kk += 4) {
                v2f a = frag_a(hl, DH, tm * 16, kk, lane);
                v2f b;
                b.x = Wm[(kk + off)     * DH + ncol];
                b.y = Wm[(kk + off + 1) * DH + ncol];
                acc = WMMA_F32(acc, a, b);
            }
            const float bsv = bm[ncol];
            const int rbase = tm * 16 + ((lane >= 16) ? 8 : 0);
            #pragma unroll
            for (int j = 0; j < 8; ++j)
                dst[(rbase + j) * DH + ncol] = acc[j] + bsv;
        }
    }
    __syncthreads();

    // ---- scores = q @ k^T / 16 : 16 tiles over 8 waves ----
    for (int t = 0; t < 2; ++t) {
        const int g = wave + t * 8;
        const int tm = g >> 2, tn = g & 3;
        v8f acc = {};
        for (int kk = 0; kk < DH; kk += 4) {
            v2f a = frag_a(ql, DH, tm * 16, kk, lane);
            v2f b = frag_a(kl, DH, tn * 16, kk, lane);   // B = k^T -> same pattern as A
            acc = WMMA_F32(acc, a, b);
        }
        const int col = tn * 16 + (lane & 15);
        const int rbase = tm * 16 + ((lane >= 16) ? 8 : 0);
        #pragma unroll
        for (int j = 0; j < 8; ++j)
            sc[(rbase + j) * 65 + col] = acc[j] * (1.f / 16.f);
    }
    __syncthreads();

    // ---- Sinkhorn: 50 alternating row/col logsumexp normalizations ----
    {
        const int r = tid >> 2, q4 = tid & 3;
        for (int it = 0; it < 50; ++it) {
            // row pass
            float mx = -3.4e38f;
            for (int j = 0; j < 16; ++j) mx = fmaxf(mx, sc[r * 65 + q4 * 16 + j]);
            mx = fmaxf(mx, __shfl_xor(mx, 1)); mx = fmaxf(mx, __shfl_xor(mx, 2));
            float s = 0.f;
            for (int j = 0; j < 16; ++j) s += __expf(sc[r * 65 + q4 * 16 + j] - mx);
            s += __shfl_xor(s, 1); s += __shfl_xor(s, 2);
            float lse = mx + __logf(s);
            for (int j = 0; j < 16; ++j) sc[r * 65 + q4 * 16 + j] -= lse;
            __syncthreads();
            // col pass (column r, 16 rows per lane)
            mx = -3.4e38f;
            for (int j = 0; j < 16; ++j) mx = fmaxf(mx, sc[(q4 * 16 + j) * 65 + r]);
            mx = fmaxf(mx, __shfl_xor(mx, 1)); mx = fmaxf(mx, __shfl_xor(mx, 2));
            s = 0.f;
            for (int j = 0; j < 16; ++j) s += __expf(sc[(q4 * 16 + j) * 65 + r] - mx);
            s += __shfl_xor(s, 1); s += __shfl_xor(s, 2);
            lse = mx + __logf(s);
            for (int j = 0; j < 16; ++j) sc[(q4 * 16 + j) * 65 + r] -= lse;
            __syncthreads();
        }
        // attn = exp(log_alpha)
        for (int j = 0; j < 16; ++j) sc[r * 65 + q4 * 16 + j] = __expf(sc[r * 65 + q4 * 16 + j]);
    }
    __syncthreads();

    // ---- av = attn @ v : 64 tiles over 8 waves, result into ql (q dead) ----
    for (int t = 0; t < 8; ++t) {
        const int g = wave + t * 8;
        const int tm = g >> 4, tn = g & 15;
        v8f acc = {};
        for (int kk = 0; kk < NAG; kk += 4) {
            v2f a = frag_a(sc, 65, tm * 16, kk, lane);
            v2f b = frag_b(vl, DH, tn * 16, kk, lane);
            acc = WMMA_F32(acc, a, b);
        }
        const int col = tn * 16 + (lane & 15);
        const int rbase = tm * 16 + ((lane >= 16) ? 8 : 0);
        #pragma unroll
        for (int j = 0; j < 8; ++j)
            ql[(rbase + j) * DH + col] = acc[j];
    }
    __syncthreads();

    // ---- mix, bias, scale, non-expansive bound, residual (in place on global h) ----
    {
        const int row = tid >> 2, q4 = tid & 3;
        const float sfv = sf[row];
        float ss = 0.f;
        for (int j = 0; j < 64; ++j) {
            int c = q4 * 64 + j;
            float z = (0.1f * hl[row * DH + c] + 0.9f * ql[row * DH + c] + ab[row * DH + c]) * sfv;
            ss += z * z;
        }
        ss += __shfl_xor(ss, 1); ss += __shfl_xor(ss, 2);
        const float scale = 1.f / fmaxf(1.f, sqrtf(ss));   // BOUND = 1
        for (int j = 0; j < 64; ++j) {
            int c = q4 * 64 + j;
            float z = (0.1f * hl[row * DH + c] + 0.9f * ql[row * DH + c] + ab[row * DH + c]) * sfv * scale;
            hg[row * DH + c] += z;
        }
    }
}

extern "C" void kernel_launch(void* const* d_in, const int* in_sizes, int n_in,
                              void* d_out, int out_size, void* d_ws, size_t ws_size,
                              hipStream_t stream) {
    const float* x    = (const float*)d_in[0];   // [1024,64,256]
    const float* Win  = (const float*)d_in[1];   // [256,256]
    const float* bin_ = (const float*)d_in[2];   // [256]
    const float* ln_g = (const float*)d_in[3];   // [3,256]
    const float* ln_b = (const float*)d_in[4];   // [3,256]
    const float* Wq   = (const float*)d_in[5];   // [3,256,256]
    const float* bq   = (const float*)d_in[6];   // [3,256]
    const float* Wk   = (const float*)d_in[7];
    const float* bk   = (const float*)d_in[8];
    const float* Wv   = (const float*)d_in[9];
    const float* bv   = (const float*)d_in[10];
    const float* ab   = (const float*)d_in[11];  // [3,64,256]
    const float* sf   = (const float*)d_in[12];  // [3,64]
    const float* Wout = (const float*)d_in[13];  // [256,128]
    const float* bout = (const float*)d_in[14];  // [128]
    float* out = (float*)d_out;                  // [1024,64,128]

    float* h = (float*)d_ws;                     // 64 MB: [65536, 256]
    const int Mrows = 1024 * 64;

    // h = x @ Win + bin_
    dim3 g1(Mrows / TM, 256 / TN);
    gemm_bias_kernel<<<g1, 256, 0, stream>>>(x, Win, bin_, h, 256);

    // 3 fused MHC layers, in place on h
    for (int l = 0; l < 3; ++l) {
        mhc_layer_kernel<<<1024, 256, 0, stream>>>(h,
            ln_g + l * 256, ln_b + l * 256,
            Wq + (size_t)l * 256 * 256, bq + l * 256,
            Wk + (size_t)l * 256 * 256, bk + l * 256,
            Wv + (size_t)l * 256 * 256, bv + l * 256,
            ab + (size_t)l * 64 * 256, sf + (size_t)l * 64);
    }

    // out = h @ Wout + bout
    dim3 g2(Mrows / TM, 1);
    gemm_bias_kernel<<<g2, 256, 0, stream>>>(h, Wout, bout, out, 128);
}